// ResidualGATBlock_24446953848870
// MI455X (gfx1250) — compile-verified
//
#include <hip/hip_runtime.h>
#include <hip/hip_bf16.h>

// ---------------------------------------------------------------------------
// ResidualGATBlock for MI455X (gfx1250, wave32)
//   h = x @ W                        -> V_WMMA_F32_16X16X4_F32 (fp32 WMMA)
//     W staged via GLOBAL_LOAD_ASYNC_TO_LDS_B128 (ASYNCcnt, no VGPR round-trip)
//   a_src/a_dst                      -> wave-per-node reduction (shfl_xor)
//   edge softmax (3 passes)          -> L2-resident fp32 atomics
//   scatter msg, ELU + residual      -> wave-per-edge, coalesced atomics
// ---------------------------------------------------------------------------

typedef float v2f __attribute__((ext_vector_type(2)));
typedef float v8f __attribute__((ext_vector_type(8)));

#define IN_DIM   128
#define OUT_DIM  128
#define HEADS    4
#define CH       32          // channels per head == wave32 lane count
#define NEG_SLOPE 0.2f

// ------------------------- fp32 atomic max (CAS) ---------------------------
__device__ inline void atomicMaxF(float* addr, float val) {
    unsigned int* ua = (unsigned int*)addr;
    unsigned int old = *ua;
    while (__uint_as_float(old) < val) {
        unsigned int assumed = old;
        old = atomicCAS(ua, assumed, __float_as_uint(val));
        if (old == assumed) break;
    }
}

// ------------------------- Kernel 1: h = x @ W via WMMA --------------------
// Block = 256 threads = 8 waves. Each wave: 16 rows x 128 cols output tile.
// W (128x128 fp32, 64KB) staged in LDS via async-to-LDS DMA; B-frag reads
// are bank-conflict-free (consecutive columns per 16-lane group).
__global__ __launch_bounds__(256) void gat_gemm_wmma(
    const float* __restrict__ x, const float* __restrict__ W,
    float* __restrict__ h, int n)
{
    __shared__ float Wl[IN_DIM * OUT_DIM];   // 64 KB

    {   // async staging: GLOBAL_LOAD_ASYNC_TO_LDS_B128, 16B per lane per issue
        unsigned lds_base = (unsigned)(uintptr_t)(void*)Wl;  // LDS byte offset
        const char* gbase = (const char*)W;
        for (int i = threadIdx.x; i < (IN_DIM * OUT_DIM) / 4; i += 256) {
            unsigned    ldsa = lds_base + (unsigned)i * 16u;
            const void* ga   = gbase + (size_t)i * 16u;
            asm volatile("global_load_async_to_lds_b128 %0, %1, off"
                         :: "v"(ldsa), "v"(ga) : "memory");
        }
        asm volatile("s_wait_asynccnt 0x0" ::: "memory");
    }
    __syncthreads();

    const int lane = threadIdx.x & 31;
    const int wave = threadIdx.x >> 5;
    const int lo   = lane & 15;         // M (A) / N (B,C,D) index
    const int hi   = lane >> 4;         // K-half selector
    const int m0   = blockIdx.x * 128 + wave * 16;

    int row = m0 + lo;                  // clamp OOB rows for loads only
    if (row >= n) row = n - 1;
    const float* xrow = x + (size_t)row * IN_DIM;

    v8f acc[8] = {};                    // 8 N-tiles of 16x16 fp32

    for (int k = 0; k < IN_DIM; k += 4) {
        // A frag: lanes 0-15 -> K = k,k+1 ; lanes 16-31 -> K = k+2,k+3
        v2f a = *(const v2f*)(xrow + k + 2 * hi);
        #pragma unroll
        for (int j = 0; j < 8; ++j) {
            v2f b;
            b.x = Wl[(k     + 2 * hi) * OUT_DIM + j * 16 + lo];
            b.y = Wl[(k + 1 + 2 * hi) * OUT_DIM + j * 16 + lo];
            acc[j] = __builtin_amdgcn_wmma_f32_16x16x4_f32(
                false, a, false, b, (short)0, acc[j], false, false);
        }
    }

    // D layout: VGPR r -> M = m0 + r (lanes 0-15) / m0 + r + 8 (lanes 16-31)
    #pragma unroll
    for (int j = 0; j < 8; ++j) {
        #pragma unroll
        for (int r = 0; r < 8; ++r) {
            int rr = m0 + r + 8 * hi;
            if (rr < n)
                h[(size_t)rr * OUT_DIM + j * 16 + lo] = acc[j][r];
        }
    }
}

// ------------------------- Kernel 2: init scratch --------------------------
__global__ void gat_init(float* __restrict__ accum, float* __restrict__ mbuf,
                         float* __restrict__ denom, int n)
{
    int i = blockIdx.x * blockDim.x + threadIdx.x;
    if (i < n * OUT_DIM) accum[i] = 0.f;
    if (i < n * HEADS) { mbuf[i] = -3.0e38f; denom[i] = 0.f; }
}

// ------------------- Kernel 3: a_src/a_dst (wave per node) -----------------
__global__ __launch_bounds__(256) void gat_attn_coef(
    const float* __restrict__ h, const float* __restrict__ att_src,
    const float* __restrict__ att_dst, float* __restrict__ a_src,
    float* __restrict__ a_dst, int n)
{
    const int lane = threadIdx.x & 31;
    const int node = blockIdx.x * 8 + (threadIdx.x >> 5);
    if (node >= n) return;
    const float* hp = h + (size_t)node * OUT_DIM;
    #pragma unroll
    for (int hd = 0; hd < HEADS; ++hd) {
        float v = hp[hd * CH + lane];
        float s = v * att_src[hd * CH + lane];
        float d = v * att_dst[hd * CH + lane];
        #pragma unroll
        for (int off = 16; off > 0; off >>= 1) {
            s += __shfl_xor(s, off, 32);
            d += __shfl_xor(d, off, 32);
        }
        if (lane == 0) {
            a_src[node * HEADS + hd] = s;
            a_dst[node * HEADS + hd] = d;
        }
    }
}

// -------------- Kernel 4: edge logits + segment max (pass 1) ---------------
__global__ void gat_edge_logits(
    const long long* __restrict__ src, const long long* __restrict__ dst,
    const float* __restrict__ a_src, const float* __restrict__ a_dst,
    float* __restrict__ alpha, float* __restrict__ mbuf,
    float* __restrict__ out_srcdst, int E_, int tot)
{
    int e = blockIdx.x * blockDim.x + threadIdx.x;
    if (e >= tot) return;
    long long s, d;
    if (e < E_) { s = src[e]; d = dst[e]; } else { s = d = (long long)(e - E_); }
    out_srcdst[e]       = (float)s;       // stacked [src; dst] output
    out_srcdst[tot + e] = (float)d;
    #pragma unroll
    for (int hd = 0; hd < HEADS; ++hd) {
        float al = a_src[s * HEADS + hd] + a_dst[d * HEADS + hd];
        al = (al > 0.f) ? al : NEG_SLOPE * al;       // leaky_relu
        alpha[(size_t)e * HEADS + hd] = al;
        atomicMaxF(&mbuf[d * HEADS + hd], al);
    }
}

// -------------- Kernel 5: exp + segment sum (pass 2) -----------------------
__global__ void gat_edge_exp(
    const long long* __restrict__ dst, float* __restrict__ alpha,
    const float* __restrict__ mbuf, float* __restrict__ denom, int E_, int tot)
{
    int e = blockIdx.x * blockDim.x + threadIdx.x;
    if (e >= tot) return;
    long long d = (e < E_) ? dst[e] : (long long)(e - E_);
    #pragma unroll
    for (int hd = 0; hd < HEADS; ++hd) {
        float ex = __expf(alpha[(size_t)e * HEADS + hd] - mbuf[d * HEADS + hd]);
        alpha[(size_t)e * HEADS + hd] = ex;
        atomicAdd(&denom[d * HEADS + hd], ex);
    }
}

// ------- Kernel 6: normalize + scatter messages (wave per edge) ------------
// lane = channel-within-head (==32, exact wave32 fit): fully coalesced
// 128B gathers of h[src] (L2-resident) and coalesced fp32 atomic adds.
__global__ __launch_bounds__(256) void gat_scatter(
    const long long* __restrict__ src, const long long* __restrict__ dst,
    const float* __restrict__ h, const float* __restrict__ alphaExp,
    const float* __restrict__ denom, float* __restrict__ accum,
    float* __restrict__ out_alpha, int E_, int tot)
{
    const int lane = threadIdx.x & 31;
    const int e = blockIdx.x * 8 + (threadIdx.x >> 5);
    if (e >= tot) return;
    long long s, d;
    if (e < E_) { s = src[e]; d = dst[e]; } else { s = d = (long long)(e - E_); }

    const float* hs = h + (size_t)s * OUT_DIM;
    float*       ad = accum + (size_t)d * OUT_DIM;
    __builtin_prefetch(hs, 0, 0);                 // global_prefetch_b8

    if (lane < HEADS) {
        float a = alphaExp[(size_t)e * HEADS + lane] /
                  fmaxf(denom[d * HEADS + lane], 1e-16f);
        out_alpha[(size_t)e * HEADS + lane] = a;
    }
    #pragma unroll
    for (int hd = 0; hd < HEADS; ++hd) {
        float a = alphaExp[(size_t)e * HEADS + hd] /
                  fmaxf(denom[d * HEADS + hd], 1e-16f);
        atomicAdd(&ad[hd * CH + lane], hs[hd * CH + lane] * a);
    }
}

// ------------- Kernel 7: bias + ELU + residual -----------------------------
__global__ void gat_finalize(
    const float* __restrict__ accum, const float* __restrict__ bias,
    const float* __restrict__ x, float* __restrict__ out, int total)
{
    int i = blockIdx.x * blockDim.x + threadIdx.x;
    if (i >= total) return;
    float v = accum[i] + bias[i & (OUT_DIM - 1)];
    v = (v > 0.f) ? v : (__expf(v) - 1.f);        // elu
    out[i] = v + x[i];
}

// ---------------------------------------------------------------------------
extern "C" void kernel_launch(void* const* d_in, const int* in_sizes, int n_in,
                              void* d_out, int out_size, void* d_ws, size_t ws_size,
                              hipStream_t stream) {
    const float*     x       = (const float*)d_in[0];
    const long long* ei      = (const long long*)d_in[1];   // [2, E] int64
    const float*     W       = (const float*)d_in[2];
    const float*     att_src = (const float*)d_in[3];
    const float*     att_dst = (const float*)d_in[4];
    const float*     bias    = (const float*)d_in[5];

    const int n   = in_sizes[0] / IN_DIM;
    const int E_  = in_sizes[1] / 2;
    const int tot = E_ + n;                                  // with self-loops

    float* out = (float*)d_out;
    // output layout: out [n*128] | stacked src,dst [2*tot] | alpha [tot*4]
    float* out_srcdst = out + (size_t)n * OUT_DIM;
    float* out_alpha  = out_srcdst + (size_t)2 * tot;

    // workspace layout (floats)
    float* ws     = (float*)d_ws;
    float* h      = ws;                                      // n*128
    float* a_src  = h      + (size_t)n * OUT_DIM;            // n*4
    float* a_dst  = a_src  + (size_t)n * HEADS;              // n*4
    float* mbuf   = a_dst  + (size_t)n * HEADS;              // n*4
    float* denom  = mbuf   + (size_t)n * HEADS;              // n*4
    float* alpha  = denom  + (size_t)n * HEADS;              // tot*4
    float* accum  = alpha  + (size_t)tot * HEADS;            // n*128

    const long long* srcI = ei;
    const long long* dstI = ei + E_;

    gat_gemm_wmma<<<(n + 127) / 128, 256, 0, stream>>>(x, W, h, n);
    gat_init<<<(n * OUT_DIM + 255) / 256, 256, 0, stream>>>(accum, mbuf, denom, n);
    gat_attn_coef<<<(n + 7) / 8, 256, 0, stream>>>(h, att_src, att_dst,
                                                   a_src, a_dst, n);
    gat_edge_logits<<<(tot + 255) / 256, 256, 0, stream>>>(
        srcI, dstI, a_src, a_dst, alpha, mbuf, out_srcdst, E_, tot);
    gat_edge_exp<<<(tot + 255) / 256, 256, 0, stream>>>(
        dstI, alpha, mbuf, denom, E_, tot);
    gat_scatter<<<(tot + 7) / 8, 256, 0, stream>>>(
        srcI, dstI, h, alpha, denom, accum, out_alpha, E_, tot);
    gat_finalize<<<(n * OUT_DIM + 255) / 256, 256, 0, stream>>>(
        accum, bias, x, out, n * OUT_DIM);
}